// Triplets_18562848654091
// MI455X (gfx1250) — compile-verified
//
#include <hip/hip_runtime.h>

#define D_MAX      64
#define N_ATOMS_FX 20000
#define BS         256
#define CUT2       2.25f

typedef int v4i __attribute__((ext_vector_type(4)));

// ---------------------------------------------------------------------------
// K1: zero per-node counters (workspace is poisoned, must init every call).
// ---------------------------------------------------------------------------
__global__ void k_zero(int* __restrict__ counts, int n) {
    int i = blockIdx.x * blockDim.x + threadIdx.x;
    if (i < n) counts[i] = 0;
}

// ---------------------------------------------------------------------------
// K2: per-edge cutoff test + scatter edge id into its source node's slot
// table. rowv[e] = source node if valid else -1. Slot order from atomicAdd is
// nondeterministic, but K3 sorts each list, restoring determinism.
// Validity uses explicit rn mul/add (no fma contraction) so the association
// matches x*x + y*y + z*z; with correctly-rounded sqrt, (||v|| <= 1.5) is
// exactly (sumsq <= 2.25) since both constants are exact in binary.
// ---------------------------------------------------------------------------
__global__ void k_build(const float* __restrict__ vec,
                        const int*   __restrict__ row,
                        int* __restrict__ rowv,
                        int* __restrict__ counts,
                        int* __restrict__ node_edges,
                        int E) {
    int e = blockIdx.x * blockDim.x + threadIdx.x;
    if (e >= E) return;
    // stream-ahead prefetch (global_prefetch_b8); speculative, OOB is dropped
    __builtin_prefetch(&vec[3 * (e + BS)], 0, 1);
    float x = vec[3 * e + 0];
    float y = vec[3 * e + 1];
    float z = vec[3 * e + 2];
    float s = __fadd_rn(__fadd_rn(__fmul_rn(x, x), __fmul_rn(y, y)),
                        __fmul_rn(z, z));
    int rv = -1;
    if (s <= CUT2) {
        int r = row[e];
        rv = r;
        int slot = atomicAdd(&counts[r], 1);
        if (slot < D_MAX) node_edges[r * D_MAX + slot] = e;
    }
    rowv[e] = rv;
}

// ---------------------------------------------------------------------------
// K3: clamp each node's count to D_MAX and insertion-sort its edge list
// ascending (typical length ~8). Ascending edge ids == the reference's
// stable-argsort slot order.
// ---------------------------------------------------------------------------
__global__ void k_sort(int* __restrict__ counts,
                       int* __restrict__ node_edges,
                       int n_atoms) {
    int r = blockIdx.x * blockDim.x + threadIdx.x;
    if (r >= n_atoms) return;
    int n = counts[r];
    if (n > D_MAX) { n = D_MAX; counts[r] = n; }
    int* lst = node_edges + r * D_MAX;
    for (int i = 1; i < n; ++i) {
        int key = lst[i];
        int j = i - 1;
        while (j >= 0 && lst[j] > key) { lst[j + 1] = lst[j]; --j; }
        lst[j + 1] = key;
    }
}

// ---------------------------------------------------------------------------
// K4: block-level exclusive scan (LDS Hillis-Steele) of per-edge pair counts
// pc[e] = valid ? counts[row[e]] - 1 : 0. Writes per-element exclusive
// prefix and the per-block total.
// ---------------------------------------------------------------------------
__global__ void k_scanA(const int* __restrict__ rowv,
                        const int* __restrict__ counts,
                        int* __restrict__ offs,
                        int* __restrict__ blockSums,
                        int E) {
    __shared__ int sh[BS];
    int tid = threadIdx.x;
    int e = blockIdx.x * BS + tid;
    int v = 0;
    if (e < E) {
        int r = rowv[e];
        if (r >= 0) v = counts[r] - 1;
    }
    sh[tid] = v;
    __syncthreads();
    for (int off = 1; off < BS; off <<= 1) {
        int t = (tid >= off) ? sh[tid - off] : 0;
        __syncthreads();
        sh[tid] += t;
        __syncthreads();
    }
    int incl = sh[tid];
    if (e < E) offs[e] = incl - v;           // exclusive prefix within block
    if (tid == BS - 1) blockSums[blockIdx.x] = incl;
}

// ---------------------------------------------------------------------------
// K5: sequential exclusive scan of the block sums (nb ~ 1250; trivially fast
// and deterministic). Also writes the grand total to blockSums[nb] so the
// tail-fill kernel knows where live pairs end.
// ---------------------------------------------------------------------------
__global__ void k_scanB(int* __restrict__ blockSums, int nb) {
    if (blockIdx.x == 0 && threadIdx.x == 0) {
        int run = 0;
        for (int i = 0; i < nb; ++i) {
            int t = blockSums[i];
            blockSums[i] = run;
            run += t;
        }
        blockSums[nb] = run;                 // total pair count
    }
}

// ---------------------------------------------------------------------------
// K6: fill ONLY the dead tail [total, cap) of both output arrays with -1,
// using non-temporal 128-bit stores (the live region [0, total) is written
// exactly once, by k_emit). Chunks straddling `total` fall back to scalar
// stores. Together with k_emit this covers every output element each call.
// ---------------------------------------------------------------------------
__global__ void k_fill_tail(int* __restrict__ out,
                            const int* __restrict__ totalp,
                            int cap) {
    int total = *totalp;
    if (total > cap) total = cap;
    int n4 = cap >> 2;                       // cap is a multiple of 4 (4M)
    v4i v = { -1, -1, -1, -1 };
    int stride = gridDim.x * blockDim.x;
    for (int j = blockIdx.x * blockDim.x + threadIdx.x; j < n4; j += stride) {
        int base = j << 2;
        if (base >= total) {                 // fully dead chunk: vector NT
            __builtin_nontemporal_store(v, (v4i*)out + j);
            __builtin_nontemporal_store(v, (v4i*)(out + cap) + j);
        } else if (base + 4 > total) {       // straddle chunk: scalar tail
            for (int t = total; t < base + 4; ++t) {
                out[t]       = -1;
                out[cap + t] = -1;
            }
        }
    }
}

// ---------------------------------------------------------------------------
// K7: emit pairs. Same gridding as K4 so blockSums[blockIdx.x] is this
// block's global offset. For each valid edge e, walk its node's ascending
// list, skip self, write (e, b) at base+k. Clipped at trip_cap; k capped at
// n-1 reserved slots (only matters in the astronomically unlikely overflow
// case where e itself was dropped from a >64-degree list).
// ---------------------------------------------------------------------------
__global__ void k_emit(const int* __restrict__ rowv,
                       const int* __restrict__ counts,
                       const int* __restrict__ node_edges,
                       const int* __restrict__ offs,
                       const int* __restrict__ blockSums,
                       int* __restrict__ out,
                       int E, int trip_cap) {
    int e = blockIdx.x * BS + threadIdx.x;
    if (e >= E) return;
    int r = rowv[e];
    if (r < 0) return;
    int base  = offs[e] + blockSums[blockIdx.x];
    int n     = counts[r];
    int limit = n - 1;                 // slots reserved for this edge
    const int* lst = node_edges + r * D_MAX;
    int k = 0;
    for (int i = 0; i < n && k < limit; ++i) {
        int b = lst[i];
        if (b == e) continue;
        int pos = base + k;
        if (pos < trip_cap) {
            out[pos]            = e;   // idx_ij
            out[trip_cap + pos] = b;   // idx_kj
        }
        ++k;
    }
}

// ---------------------------------------------------------------------------
extern "C" void kernel_launch(void* const* d_in, const int* in_sizes, int n_in,
                              void* d_out, int out_size, void* d_ws, size_t ws_size,
                              hipStream_t stream) {
    const float* vec = (const float*)d_in[0];
    const int*   row = (const int*)d_in[1];   // edge_index[0] = first E entries
    (void)n_in; (void)ws_size;

    const int E        = in_sizes[0] / 3;
    const int trip_cap = out_size / 2;
    const int n_atoms  = N_ATOMS_FX;          // fixed by the reference problem

    int* out = (int*)d_out;

    // workspace carve-up (ints): counts | node_edges | rowv | offs | blockSums
    int* counts     = (int*)d_ws;
    int* node_edges = counts + n_atoms;
    int* rowv       = node_edges + (size_t)n_atoms * D_MAX;
    int* offs       = rowv + E;
    int* blockSums  = offs + E;               // nbE + 1 slots (last = total)

    const int nbE = (E + BS - 1) / BS;
    const int nbN = (n_atoms + BS - 1) / BS;

    k_zero     <<<nbN, BS, 0, stream>>>(counts, n_atoms);
    k_build    <<<nbE, BS, 0, stream>>>(vec, row, rowv, counts, node_edges, E);
    k_sort     <<<nbN, BS, 0, stream>>>(counts, node_edges, n_atoms);
    k_scanA    <<<nbE, BS, 0, stream>>>(rowv, counts, offs, blockSums, E);
    k_scanB    <<<1,   32, 0, stream>>>(blockSums, nbE);
    k_fill_tail<<<512, BS, 0, stream>>>(out, blockSums + nbE, trip_cap);
    k_emit     <<<nbE, BS, 0, stream>>>(rowv, counts, node_edges, offs,
                                        blockSums, out, E, trip_cap);
}